// OLS_4294967296554
// MI455X (gfx1250) — compile-verified
//
#include <hip/hip_runtime.h>
#include <hip/hip_bf16.h>

// ---------------------------------------------------------------------------
// NNLS via normal equations on MI455X (gfx1250).
// Phase 1: Ghat = Ahat^T Ahat with Ahat = [X | 1 | y | 0-pad] (N x 48),
//          done as rank-4 updates with V_WMMA_F32_16X16X4_F32 (3x3 tiles).
// Phase 2: power iteration (50) + FISTA (400) on the 33x33 / 33x4 problem,
//          single workgroup.
// ---------------------------------------------------------------------------

typedef __attribute__((ext_vector_type(2))) float v2f;
typedef __attribute__((ext_vector_type(8))) float v8f;

#define GDIM 48           // padded Gram dimension (3 x 16)
#define GELEMS (GDIM*GDIM)

__global__ void zero_ws_kernel(float* __restrict__ g) {
    for (int i = threadIdx.x; i < GELEMS; i += blockDim.x) g[i] = 0.0f;
}

// ---------------------------------------------------------------------------
// Gram accumulation kernel.
// Each wave processes 4-row chunks of Ahat.  Per chunk it builds 3 fragments
// (cols 0-15, 16-31, 32-47 of Ahat).  Because the f32 WMMA A-fragment
// (16x4, M=lane%16, K=2*(lane/16)+vgpr) and B-fragment (4x16, N=lane%16,
// K=vgpr+2*(lane/16)) index Ahat identically for a Gram product, the same
// registers serve as both operands: acc[i][j] += frag[i]^T-ish * frag[j].
// ---------------------------------------------------------------------------
__global__ void gram_kernel(const float* __restrict__ X,
                            const float* __restrict__ Yv,
                            float* __restrict__ Gws,
                            int nsteps) {
    const int lane  = threadIdx.x & 31;
    const int half  = lane >> 4;      // 0: K=0,1   1: K=2,3
    const int c     = lane & 15;      // column within 16-wide tile
    const int wavesPerBlk = blockDim.x >> 5;
    const int wave   = blockIdx.x * wavesPerBlk + (threadIdx.x >> 5);
    const int nwaves = gridDim.x * wavesPerBlk;

    v8f acc[3][3];
#pragma unroll
    for (int i = 0; i < 3; ++i)
#pragma unroll
        for (int j = 0; j < 3; ++j)
#pragma unroll
            for (int e = 0; e < 8; ++e) acc[i][j][e] = 0.0f;

    for (int s = wave; s < nsteps; s += nwaves) {
        const long row0 = 4L * s + 2 * half;   // K = 2*half + 0
        const long row1 = row0 + 1;            // K = 2*half + 1
        const float* __restrict__ x0 = X + row0 * 32;
        const float* __restrict__ x1 = X + row1 * 32;

        v2f fr[3];
        // fragment 0: Ahat cols 0..15  (X cols 0..15)
        fr[0].x = x0[c];
        fr[0].y = x1[c];
        // fragment 1: Ahat cols 16..31 (X cols 16..31)
        fr[1].x = x0[16 + c];
        fr[1].y = x1[16 + c];
        // fragment 2: Ahat cols 32..47  (col32 = 1, col33..36 = y, rest 0)
        float e0, e1;
        if (c == 0)      { e0 = 1.0f; e1 = 1.0f; }
        else if (c <= 4) { e0 = Yv[row0 * 4 + (c - 1)];
                           e1 = Yv[row1 * 4 + (c - 1)]; }
        else             { e0 = 0.0f; e1 = 0.0f; }
        fr[2].x = e0;
        fr[2].y = e1;

#pragma unroll
        for (int i = 0; i < 3; ++i)
#pragma unroll
            for (int j = 0; j < 3; ++j)
                acc[i][j] = __builtin_amdgcn_wmma_f32_16x16x4_f32(
                    false, fr[i], false, fr[j], (short)0, acc[i][j],
                    false, false);
    }

    // Reduce partial Gram tiles into global workspace.
    // C/D layout: VGPR r, lane L -> (M = r + 8*(L/16), N = L%16).
#pragma unroll
    for (int i = 0; i < 3; ++i)
#pragma unroll
        for (int j = 0; j < 3; ++j)
#pragma unroll
            for (int r = 0; r < 8; ++r) {
                const int gm = 16 * i + r + 8 * half;
                const int gn = 16 * j + c;
                atomicAdd(&Gws[gm * GDIM + gn], acc[i][j][r]);
            }
}

// ---------------------------------------------------------------------------
// Solver: power iteration for L = lambda_max(G), then FISTA.
// G = Ghat[0:33, 0:33], c = Ghat[0:33, 33:37].
// 132 active threads, one per (m, n) element of Z (33 x 4).
// Each thread keeps its G row in registers.
// ---------------------------------------------------------------------------
#define DD 33
#define POWER_ITERS 50
#define QP_ITERS 400

__global__ void solver_kernel(const float* __restrict__ G48,
                              float* __restrict__ out) {
    __shared__ float Ys[DD * 4];
    __shared__ float Zs[DD * 4];
    __shared__ float sv[DD];
    __shared__ float red[DD];

    const int tid = threadIdx.x;
    const int m = tid >> 2;
    const int n = tid & 3;
    const bool act = (tid < DD * 4);
    const bool pw  = act && (n == 0);   // power-iteration lanes, one per row m

    float g[DD];
    float cmn = 0.0f;
    if (act) {
#pragma unroll
        for (int k = 0; k < DD; ++k) g[k] = G48[m * GDIM + k];
        cmn = G48[m * GDIM + 33 + n];
    }

    // --- power iteration: v <- G v / ||G v||, 50 times -------------------
    if (pw) sv[m] = rsqrtf((float)DD);   // v0 = 1/sqrt(33)
    __syncthreads();

    for (int it = 0; it < POWER_ITERS; ++it) {
        float w = 0.0f;
        if (pw) {
#pragma unroll
            for (int k = 0; k < DD; ++k) w += g[k] * sv[k];
            red[m] = w * w;
        }
        __syncthreads();
        if (pw) {
            float s = 0.0f;
#pragma unroll
            for (int k = 0; k < DD; ++k) s += red[k];
            sv[m] = w * rsqrtf(s);
        }
        __syncthreads();
    }

    // --- L = v^T G v ------------------------------------------------------
    if (pw) {
        float w = 0.0f;
#pragma unroll
        for (int k = 0; k < DD; ++k) w += g[k] * sv[k];
        red[m] = w * sv[m];
    }
    __syncthreads();
    float L = 0.0f;
#pragma unroll
    for (int k = 0; k < DD; ++k) L += red[k];
    const float step = 1.0f / L;

    // --- FISTA ------------------------------------------------------------
    if (act) { Ys[tid] = 0.0f; Zs[tid] = 0.0f; }
    __syncthreads();

    float t = 1.0f;
    for (int it = 0; it < QP_ITERS; ++it) {
        float zn = 0.0f;
        if (act) {
            float gy = 0.0f;
#pragma unroll
            for (int k = 0; k < DD; ++k) gy += g[k] * Ys[k * 4 + n];
            zn = Ys[tid] - step * (gy - cmn);
            if (m < 32) zn = fmaxf(zn, 0.0f);     // bias row (m==32) is free
        }
        const float tn = 0.5f * (1.0f + sqrtf(1.0f + 4.0f * t * t));
        const float beta = (t - 1.0f) / tn;
        __syncthreads();                 // all reads of Ys done
        if (act) {
            const float zo = Zs[tid];
            Ys[tid] = zn + beta * (zn - zo);
            Zs[tid] = zn;
        }
        t = tn;
        __syncthreads();
    }

    if (act && m < 32) out[tid] = Zs[tid];   // W is rows 0..31 -> 128 floats
}

// ---------------------------------------------------------------------------
extern "C" void kernel_launch(void* const* d_in, const int* in_sizes, int n_in,
                              void* d_out, int out_size, void* d_ws, size_t ws_size,
                              hipStream_t stream) {
    const float* X  = (const float*)d_in[0];   // [N, 32] f32
    const float* Yv = (const float*)d_in[1];   // [N, 4]  f32
    float* out = (float*)d_out;                // [32, 4] f32
    float* Gws = (float*)d_ws;                 // 48*48 f32 accumulator

    const int N = in_sizes[0] / 32;
    const int nsteps = N / 4;

    zero_ws_kernel<<<dim3(1), dim3(256), 0, stream>>>(Gws);
    gram_kernel<<<dim3(1024), dim3(256), 0, stream>>>(X, Yv, Gws, nsteps);
    solver_kernel<<<dim3(1), dim3(160), 0, stream>>>(Gws, out);
}